// GNN_62423054680132
// MI455X (gfx1250) — compile-verified
//
#include <hip/hip_runtime.h>

typedef __attribute__((ext_vector_type(2))) float v2f;
typedef __attribute__((ext_vector_type(8))) float v8f;

#define N_NODES   100000
#define N_EDGES   800000
#define N_ETYPES  5
#define IN_DIM    23
#define IN_PAD    24
#define HID_DIM   128
#define OUT_DIM   64
#define NUM_GRAPHS 64

// ---------------- utility zero kernels ----------------
__global__ void k_zero_i32(int* p, int n) {
    int i = blockIdx.x * blockDim.x + threadIdx.x;
    if (i < n) p[i] = 0;
}
__global__ void k_zero_f32(float* p, int n) {
    int i = blockIdx.x * blockDim.x + threadIdx.x;
    if (i < n) p[i] = 0.f;
}

// ---------------- CSR build ----------------
__global__ void k_count_deg(const int* __restrict__ dst, int* __restrict__ deg) {
    int idx = blockIdx.x * blockDim.x + threadIdx.x;
    if (idx >= N_ETYPES * N_EDGES) return;
    int e = idx / N_EDGES;
    atomicAdd(&deg[e * N_NODES + dst[idx]], 1);
}

// one block (1024 threads) per edge type: exclusive prefix sum of degrees
__global__ void k_scan(const int* __restrict__ deg, int* __restrict__ offs) {
    __shared__ int tmp[1024];
    int e = blockIdx.x, t = threadIdx.x;
    const int* d = deg + (size_t)e * N_NODES;
    int* o = offs + (size_t)e * (N_NODES + 1);
    if (t == 0) o[0] = 0;
    int run = 0;
    for (int base = 0; base < N_NODES; base += 1024) {
        int i = base + t;
        tmp[t] = (i < N_NODES) ? d[i] : 0;
        __syncthreads();
        for (int off = 1; off < 1024; off <<= 1) {
            int x = (t >= off) ? tmp[t - off] : 0;
            __syncthreads();
            tmp[t] += x;
            __syncthreads();
        }
        if (i < N_NODES) o[i + 1] = run + tmp[t];
        run += tmp[1023];
        __syncthreads();
    }
}

__global__ void k_fill_csr(const int* __restrict__ src, const int* __restrict__ dst,
                           const int* __restrict__ offs, int* __restrict__ cursor,
                           int* __restrict__ csr) {
    int idx = blockIdx.x * blockDim.x + threadIdx.x;
    if (idx >= N_ETYPES * N_EDGES) return;
    int e = idx / N_EDGES;
    int dn = dst[idx];
    int pos = atomicAdd(&cursor[e * N_NODES + dn], 1);
    csr[(size_t)e * N_EDGES + offs[e * (N_NODES + 1) + dn] + pos] = src[idx];
}

// ---------------- GCN aggregation: h_neigh = (sum_nbrs + self) / (deg+1) ----------------
// layer-1 input features (23-wide, write padded 24-wide A); one wave per node
__global__ void k_agg_in(const float* __restrict__ feat, const int* __restrict__ csr,
                         const int* __restrict__ offs, float* __restrict__ A, int e) {
    int node = blockIdx.x * 8 + (threadIdx.x >> 5);
    if (node >= N_NODES) return;
    int lane = threadIdx.x & 31;
    const int* o = offs + (size_t)e * (N_NODES + 1);
    int beg = o[node], end = o[node + 1];
    bool act = lane < IN_DIM;
    float acc = act ? feat[(size_t)node * IN_DIM + lane] : 0.f;
    const int* cs = csr + (size_t)e * N_EDGES;
    for (int j = beg; j < end; ++j) {
        int nb = cs[j];
        if (act) acc += feat[(size_t)nb * IN_DIM + lane];
    }
    float inv = 1.f / (float)(end - beg + 1);
    if (lane < IN_PAD) A[(size_t)node * IN_PAD + lane] = act ? acc * inv : 0.f;
}

// layer-2: 128-wide rows, one wave per node, float4 per lane (gathers hit L2: h1 = 51MB)
__global__ void k_agg_hid(const float* __restrict__ h, const int* __restrict__ csr,
                          const int* __restrict__ offs, float* __restrict__ A, int e) {
    int node = blockIdx.x * 8 + (threadIdx.x >> 5);
    if (node >= N_NODES) return;
    int lane = threadIdx.x & 31;
    const int* o = offs + (size_t)e * (N_NODES + 1);
    int beg = o[node], end = o[node + 1];
    float4 acc = ((const float4*)(h + (size_t)node * HID_DIM))[lane];
    const int* cs = csr + (size_t)e * N_EDGES;
    for (int j = beg; j < end; ++j) {
        int nb = cs[j];
        float4 v = ((const float4*)(h + (size_t)nb * HID_DIM))[lane];
        acc.x += v.x; acc.y += v.y; acc.z += v.z; acc.w += v.w;
    }
    float inv = 1.f / (float)(end - beg + 1);
    acc.x *= inv; acc.y *= inv; acc.z *= inv; acc.w *= inv;
    ((float4*)(A + (size_t)node * HID_DIM))[lane] = acc;
}

// ---------------- WMMA f32 GEMM: Out[16 x 16*CTILES strip] (+)= A @ B + bias ----------------
// One wave computes CTILES adjacent 16x16 tiles of one row strip:
//   - A fragment loaded once per k-step, reused across CTILES
//   - CTILES independent accumulator chains keep the matrix pipe busy
//   - LDB/LDO compile-time: B/Out accesses are immediate-offset loads from fixed
//     per-lane base pointers (no per-step 64-bit address math)
// Fragment layouts per CDNA5 ISA 7.12.2:
//   A 16x4 : lane(l16,half) holds A[r0+l16, k+half*2], A[r0+l16, k+half*2+1]
//   B 4x16 : lane holds B[k+half*2, col], B[k+half*2+1, col]
//   C/D    : VGPR r <-> row r0 + half*8 + r, col c0 + l16
template<int K, int KVALID, int LDA, int LDB, int LDO, int CTILES>
__global__ void k_gemm16(const float* __restrict__ A,
                         const float* __restrict__ B,
                         const float* __restrict__ bias,
                         float* __restrict__ Out,
                         int first, int relu) {
    constexpr bool GUARD = (KVALID != K);
    int lane = threadIdx.x & 31;
    int half = lane >> 4, l16 = lane & 15;
    int r0 = blockIdx.x * 16, c0 = blockIdx.y * (16 * CTILES);

    // fixed per-lane base pointers; all inner accesses use compile-time offsets
    const float* Ap = A + (size_t)(r0 + l16) * LDA + half * 2;          // + k
    const float* Bp = B + (size_t)(half * 2) * LDB + c0 + l16;         // + k*LDB + t*16 (+LDB)
    float*       Op = Out + (size_t)(r0 + half * 8) * LDO + c0 + l16;  // + r*LDO + t*16

    v8f acc[CTILES];
#pragma unroll
    for (int t = 0; t < CTILES; ++t) acc[t] = (v8f){};

    if (!first) {
#pragma unroll
        for (int t = 0; t < CTILES; ++t)
#pragma unroll
            for (int r = 0; r < 8; ++r)
                acc[t][r] = Op[r * LDO + t * 16];
    }

#pragma unroll
    for (int k = 0; k < K; k += 4) {
        float2 af = *(const float2*)(Ap + k);
        v2f a; a.x = af.x; a.y = af.y;
#pragma unroll
        for (int t = 0; t < CTILES; ++t) {
            v2f b;
            if (GUARD) {
                int bk = k + half * 2;
                b.x = (bk     < KVALID) ? Bp[k * LDB + t * 16]       : 0.f;
                b.y = (bk + 1 < KVALID) ? Bp[k * LDB + t * 16 + LDB] : 0.f;
            } else {
                b.x = Bp[k * LDB + t * 16];
                b.y = Bp[k * LDB + t * 16 + LDB];
            }
            acc[t] = __builtin_amdgcn_wmma_f32_16x16x4_f32(false, a, false, b,
                                                           (short)0, acc[t], false, false);
        }
    }

#pragma unroll
    for (int t = 0; t < CTILES; ++t) {
        float bv = bias[c0 + t * 16 + l16];
#pragma unroll
        for (int r = 0; r < 8; ++r) {
            float v = acc[t][r] + bv;
            if (relu) v = fmaxf(v, 0.f);
            Op[r * LDO + t * 16] = v;
        }
    }
}

// ---------------- per-graph mean pooling (graph_ids sorted) ----------------
__global__ void k_counts(const int* __restrict__ gid, float* __restrict__ counts) {
    __shared__ int c[NUM_GRAPHS];
    int t = threadIdx.x;
    if (t < NUM_GRAPHS) c[t] = 0;
    __syncthreads();
    for (int i = t; i < N_NODES; i += 1024) atomicAdd(&c[gid[i]], 1);
    __syncthreads();
    if (t < NUM_GRAPHS) counts[t] = (float)c[t];
}

__global__ void k_pool(const float* __restrict__ h2, const int* __restrict__ gid,
                       float* __restrict__ out) {
    int col = threadIdx.x;                   // 64 threads = 64 output dims
    int start = blockIdx.x * 256;
    int end = start + 256; if (end > N_NODES) end = N_NODES;
    int cur = -1; float acc = 0.f;
    for (int n = start; n < end; ++n) {
        int g = gid[n];
        if (g != cur) {
            if (cur >= 0) atomicAdd(&out[cur * OUT_DIM + col], acc);
            cur = g; acc = 0.f;
        }
        acc += h2[(size_t)n * OUT_DIM + col];
    }
    if (cur >= 0) atomicAdd(&out[cur * OUT_DIM + col], acc);
}

__global__ void k_final(float* __restrict__ out, const float* __restrict__ counts) {
    int g = blockIdx.x, c = threadIdx.x;
    out[g * OUT_DIM + c] /= fmaxf(counts[g], 1.f);
}

// ---------------- driver ----------------
extern "C" void kernel_launch(void* const* d_in, const int* in_sizes, int n_in,
                              void* d_out, int out_size, void* d_ws, size_t ws_size,
                              hipStream_t stream) {
    const float* feat = (const float*)d_in[0];
    const int*   src  = (const int*)d_in[1];
    const int*   dst  = (const int*)d_in[2];
    const int*   gid  = (const int*)d_in[3];
    const float* W1   = (const float*)d_in[4];
    const float* b1   = (const float*)d_in[5];
    const float* W2   = (const float*)d_in[6];
    const float* b2   = (const float*)d_in[7];
    float* out = (float*)d_out;

    char* p = (char*)d_ws;
    auto take = [&](size_t bytes) -> char* {
        char* r = p; p += (bytes + 255) & ~(size_t)255; return r;
    };
    float* h1     = (float*)take((size_t)N_NODES * HID_DIM * 4);
    float* h2     = (float*)take((size_t)N_NODES * OUT_DIM * 4);
    float* Abuf   = (float*)take((size_t)N_NODES * HID_DIM * 4);
    int*   deg    = (int*)take((size_t)N_ETYPES * N_NODES * 4);
    int*   offs   = (int*)take((size_t)N_ETYPES * (N_NODES + 1) * 4);
    int*   cursor = (int*)take((size_t)N_ETYPES * N_NODES * 4);
    int*   csr    = (int*)take((size_t)N_ETYPES * N_EDGES * 4);
    float* counts = (float*)take(NUM_GRAPHS * 4);

    int nz = N_ETYPES * N_NODES;
    k_zero_i32<<<(nz + 255) / 256, 256, 0, stream>>>(deg, nz);
    k_zero_i32<<<(nz + 255) / 256, 256, 0, stream>>>(cursor, nz);
    k_zero_f32<<<(NUM_GRAPHS * OUT_DIM + 255) / 256, 256, 0, stream>>>(out, NUM_GRAPHS * OUT_DIM);

    int ne = N_ETYPES * N_EDGES;
    k_count_deg<<<(ne + 255) / 256, 256, 0, stream>>>(dst, deg);
    k_scan<<<N_ETYPES, 1024, 0, stream>>>(deg, offs);
    k_fill_csr<<<(ne + 255) / 256, 256, 0, stream>>>(src, dst, offs, cursor, csr);

    dim3 g1(N_NODES / 16, HID_DIM / 64);   // 16x64 strip per wave, 2 strips across 128 cols
    dim3 g2(N_NODES / 16, OUT_DIM / 64);   // whole 64-wide output strip per wave
    int aggBlocks = (N_NODES + 7) / 8;

    // layer 1: feat(23) -> h1(128), ReLU fused on last etype
    for (int e = 0; e < N_ETYPES; ++e) {
        k_agg_in<<<aggBlocks, 256, 0, stream>>>(feat, csr, offs, Abuf, e);
        k_gemm16<IN_PAD, IN_DIM, IN_PAD, HID_DIM, HID_DIM, 4><<<g1, 32, 0, stream>>>(
            Abuf, W1 + (size_t)e * IN_DIM * HID_DIM,
            b1 + (size_t)e * HID_DIM, h1,
            e == 0 ? 1 : 0, e == N_ETYPES - 1 ? 1 : 0);
    }
    // layer 2: h1(128) -> h2(64)
    for (int e = 0; e < N_ETYPES; ++e) {
        k_agg_hid<<<aggBlocks, 256, 0, stream>>>(h1, csr, offs, Abuf, e);
        k_gemm16<HID_DIM, HID_DIM, HID_DIM, OUT_DIM, OUT_DIM, 4><<<g2, 32, 0, stream>>>(
            Abuf, W2 + (size_t)e * HID_DIM * OUT_DIM,
            b2 + (size_t)e * OUT_DIM, h2,
            e == 0 ? 1 : 0, 0);
    }

    k_counts<<<1, 1024, 0, stream>>>(gid, counts);
    k_pool<<<(N_NODES + 255) / 256, OUT_DIM, 0, stream>>>(h2, gid, out);
    k_final<<<NUM_GRAPHS, OUT_DIM, 0, stream>>>(out, counts);
}